// HeadwiseLowRankModule_7610682048912
// MI455X (gfx1250) — compile-verified
//
#include <hip/hip_runtime.h>

typedef __attribute__((ext_vector_type(2))) float v2f;
typedef __attribute__((ext_vector_type(8))) float v8f;

#define NUM_GROUPS 32
#define RANK       64
#define GROUP_DIM  128
#define IN_F       4096
#define OUT_F      4096
#define M_TILE     128
#define KT         64

// LDS row stride (floats) for 64-wide tiles, padded: bank index = 4*m + k -> conflict-free
#define STRIDE64   68

#define WMMA_F32(A, B, C) \
    __builtin_amdgcn_wmma_f32_16x16x4_f32(false, (A), false, (B), (short)0, (C), false, false)

__global__ __launch_bounds__(256)
void hlr_fused_kernel(const float* __restrict__ H,
                      const float* __restrict__ VT,
                      const float* __restrict__ U,
                      float* __restrict__ Out)
{
    __shared__ float sH[M_TILE * STRIDE64];      // 34816 B
    __shared__ float sV[RANK   * STRIDE64];      // 17408 B
    __shared__ float sU[GROUP_DIM * STRIDE64];   // 34816 B
    __shared__ float sZ[M_TILE * STRIDE64];      // 34816 B  (total ~119 KB)

    const int tid   = threadIdx.x;
    const int lane  = tid & 31;
    const int wave  = tid >> 5;          // 0..7
    const int laneM = lane & 15;         // row/col within 16
    const int laneK = (lane >> 4) << 1;  // 0 or 2 (K offset per ISA A/B layout)
    const int hiRow = (lane >> 4) << 3;  // +8 rows for upper half (C/D layout)

    const int    g     = blockIdx.x;                   // group 0..31
    const size_t mBase = (size_t)blockIdx.y * M_TILE;  // row tile base

    const int segk = (tid & 15) * 4;  // 0..60 : float4 column segment
    const int r0   = tid >> 4;        // 0..15 : base row

    // ---- stage U[g] (128 x 64) into LDS once ----
    {
        const float* Ug = U + (size_t)g * GROUP_DIM * RANK;
#pragma unroll
        for (int it = 0; it < 8; ++it) {
            const int d = r0 + it * 16;
            const float4 u = *(const float4*)(Ug + (size_t)d * RANK + segk);
            *(float4*)(&sU[d * STRIDE64 + segk]) = u;
        }
    }

    // ---- GEMM1: Z(128x64) = H_tile(128x4096) * VT_g(64x4096)^T ----
    const int wm = wave >> 1;  // 0..3 : 32-row strip
    const int wn = wave & 1;   // 0..1 : 32-col strip

    const float* Hp0 = H  + (mBase + r0) * (size_t)IN_F + segk;
    const float* Vp0 = VT + ((size_t)g * RANK + r0) * IN_F + segk;

    // prefetch registers (named scalars only: must stay in VGPRs, no stack)
    float4 h0, h1, h2, h3, h4, h5, h6, h7, v0, v1, v2, v3;
    h0 = *(const float4*)(Hp0 + (size_t)(0 * 16) * IN_F);
    h1 = *(const float4*)(Hp0 + (size_t)(1 * 16) * IN_F);
    h2 = *(const float4*)(Hp0 + (size_t)(2 * 16) * IN_F);
    h3 = *(const float4*)(Hp0 + (size_t)(3 * 16) * IN_F);
    h4 = *(const float4*)(Hp0 + (size_t)(4 * 16) * IN_F);
    h5 = *(const float4*)(Hp0 + (size_t)(5 * 16) * IN_F);
    h6 = *(const float4*)(Hp0 + (size_t)(6 * 16) * IN_F);
    h7 = *(const float4*)(Hp0 + (size_t)(7 * 16) * IN_F);
    v0 = *(const float4*)(Vp0 + (size_t)(0 * 16) * IN_F);
    v1 = *(const float4*)(Vp0 + (size_t)(1 * 16) * IN_F);
    v2 = *(const float4*)(Vp0 + (size_t)(2 * 16) * IN_F);
    v3 = *(const float4*)(Vp0 + (size_t)(3 * 16) * IN_F);

    v8f acc00 = {}, acc01 = {}, acc10 = {}, acc11 = {};

    const int aoff0 = (wm * 32 +      laneM) * STRIDE64 + laneK;
    const int aoff1 = (wm * 32 + 16 + laneM) * STRIDE64 + laneK;
    const int boff0 = (wn * 32 +      laneM) * STRIDE64 + laneK;
    const int boff1 = (wn * 32 + 16 + laneM) * STRIDE64 + laneK;

    for (int kt = 0; kt < IN_F; kt += KT) {
        __syncthreads();   // LDS slab free to overwrite
        *(float4*)(&sH[(r0 + 0 * 16) * STRIDE64 + segk]) = h0;
        *(float4*)(&sH[(r0 + 1 * 16) * STRIDE64 + segk]) = h1;
        *(float4*)(&sH[(r0 + 2 * 16) * STRIDE64 + segk]) = h2;
        *(float4*)(&sH[(r0 + 3 * 16) * STRIDE64 + segk]) = h3;
        *(float4*)(&sH[(r0 + 4 * 16) * STRIDE64 + segk]) = h4;
        *(float4*)(&sH[(r0 + 5 * 16) * STRIDE64 + segk]) = h5;
        *(float4*)(&sH[(r0 + 6 * 16) * STRIDE64 + segk]) = h6;
        *(float4*)(&sH[(r0 + 7 * 16) * STRIDE64 + segk]) = h7;
        *(float4*)(&sV[(r0 + 0 * 16) * STRIDE64 + segk]) = v0;
        *(float4*)(&sV[(r0 + 1 * 16) * STRIDE64 + segk]) = v1;
        *(float4*)(&sV[(r0 + 2 * 16) * STRIDE64 + segk]) = v2;
        *(float4*)(&sV[(r0 + 3 * 16) * STRIDE64 + segk]) = v3;
        __syncthreads();

        // prefetch next slab while computing this one
        if (kt + KT < IN_F) {
            const float* Hp = Hp0 + (kt + KT);
            const float* Vp = Vp0 + (kt + KT);
            h0 = *(const float4*)(Hp + (size_t)(0 * 16) * IN_F);
            h1 = *(const float4*)(Hp + (size_t)(1 * 16) * IN_F);
            h2 = *(const float4*)(Hp + (size_t)(2 * 16) * IN_F);
            h3 = *(const float4*)(Hp + (size_t)(3 * 16) * IN_F);
            h4 = *(const float4*)(Hp + (size_t)(4 * 16) * IN_F);
            h5 = *(const float4*)(Hp + (size_t)(5 * 16) * IN_F);
            h6 = *(const float4*)(Hp + (size_t)(6 * 16) * IN_F);
            h7 = *(const float4*)(Hp + (size_t)(7 * 16) * IN_F);
            v0 = *(const float4*)(Vp + (size_t)(0 * 16) * IN_F);
            v1 = *(const float4*)(Vp + (size_t)(1 * 16) * IN_F);
            v2 = *(const float4*)(Vp + (size_t)(2 * 16) * IN_F);
            v3 = *(const float4*)(Vp + (size_t)(3 * 16) * IN_F);
        }

#pragma unroll
        for (int kk = 0; kk < KT; kk += 4) {
            const v2f a0 = *(const v2f*)(&sH[aoff0 + kk]);
            const v2f a1 = *(const v2f*)(&sH[aoff1 + kk]);
            const v2f b0 = *(const v2f*)(&sV[boff0 + kk]);
            const v2f b1 = *(const v2f*)(&sV[boff1 + kk]);
            acc00 = WMMA_F32(a0, b0, acc00);
            acc01 = WMMA_F32(a0, b1, acc01);
            acc10 = WMMA_F32(a1, b0, acc10);
            acc11 = WMMA_F32(a1, b1, acc11);
        }
    }

    // ---- spill Z to LDS (C-layout -> A-layout transpose) ----
    {
        const int zc0 = wn * 32 +      laneM;
        const int zc1 = wn * 32 + 16 + laneM;
        const int zr0 = wm * 32 +      hiRow;   // subtile i=0
        const int zr1 = wm * 32 + 16 + hiRow;   // subtile i=1
#pragma unroll
        for (int v = 0; v < 8; ++v) {
            sZ[(zr0 + v) * STRIDE64 + zc0] = acc00[v];
            sZ[(zr0 + v) * STRIDE64 + zc1] = acc01[v];
            sZ[(zr1 + v) * STRIDE64 + zc0] = acc10[v];
            sZ[(zr1 + v) * STRIDE64 + zc1] = acc11[v];
        }
    }
    __syncthreads();

    // ---- GEMM2: Out_tile(128x128) = Z(128x64) * U_g(128x64)^T ----
    v8f c00 = {}, c01 = {}, c02 = {}, c03 = {};
    v8f c10 = {}, c11 = {}, c12 = {}, c13 = {};

    const int zoff0 = (wm * 32 +      laneM) * STRIDE64 + laneK;
    const int zoff1 = (wm * 32 + 16 + laneM) * STRIDE64 + laneK;
    const int uo0 = (wn * 64 +  0 + laneM) * STRIDE64 + laneK;
    const int uo1 = (wn * 64 + 16 + laneM) * STRIDE64 + laneK;
    const int uo2 = (wn * 64 + 32 + laneM) * STRIDE64 + laneK;
    const int uo3 = (wn * 64 + 48 + laneM) * STRIDE64 + laneK;

#pragma unroll
    for (int r = 0; r < RANK; r += 4) {
        const v2f a0 = *(const v2f*)(&sZ[zoff0 + r]);
        const v2f a1 = *(const v2f*)(&sZ[zoff1 + r]);
        const v2f b0 = *(const v2f*)(&sU[uo0 + r]);
        const v2f b1 = *(const v2f*)(&sU[uo1 + r]);
        const v2f b2 = *(const v2f*)(&sU[uo2 + r]);
        const v2f b3 = *(const v2f*)(&sU[uo3 + r]);
        c00 = WMMA_F32(a0, b0, c00);
        c01 = WMMA_F32(a0, b1, c01);
        c02 = WMMA_F32(a0, b2, c02);
        c03 = WMMA_F32(a0, b3, c03);
        c10 = WMMA_F32(a1, b0, c10);
        c11 = WMMA_F32(a1, b1, c11);
        c12 = WMMA_F32(a1, b2, c12);
        c13 = WMMA_F32(a1, b3, c13);
    }

    // ---- write output tile (streaming: non-temporal, keep L2 for H/VT) ----
    {
        float* Op = Out + mBase * OUT_F + (size_t)g * GROUP_DIM;
        const int oc0 = wn * 64 +  0 + laneM;
        const int oc1 = wn * 64 + 16 + laneM;
        const int oc2 = wn * 64 + 32 + laneM;
        const int oc3 = wn * 64 + 48 + laneM;
        const int or0 = wm * 32 +      hiRow;
        const int or1 = wm * 32 + 16 + hiRow;
#pragma unroll
        for (int v = 0; v < 8; ++v) {
            float* R0 = Op + (size_t)(or0 + v) * OUT_F;
            float* R1 = Op + (size_t)(or1 + v) * OUT_F;
            __builtin_nontemporal_store(c00[v], &R0[oc0]);
            __builtin_nontemporal_store(c01[v], &R0[oc1]);
            __builtin_nontemporal_store(c02[v], &R0[oc2]);
            __builtin_nontemporal_store(c03[v], &R0[oc3]);
            __builtin_nontemporal_store(c10[v], &R1[oc0]);
            __builtin_nontemporal_store(c11[v], &R1[oc1]);
            __builtin_nontemporal_store(c12[v], &R1[oc2]);
            __builtin_nontemporal_store(c13[v], &R1[oc3]);
        }
    }
}

extern "C" void kernel_launch(void* const* d_in, const int* in_sizes, int n_in,
                              void* d_out, int out_size, void* d_ws, size_t ws_size,
                              hipStream_t stream) {
    const float* H  = (const float*)d_in[0];  // [4,4096,4096] fp32
    const float* VT = (const float*)d_in[1];  // [2048,4096]  fp32
    const float* U  = (const float*)d_in[2];  // [32,128,64]  fp32
    float* Out = (float*)d_out;               // [4,4096,4096] fp32

    const int M = in_sizes[0] / IN_F;         // 16384 rows
    dim3 grid(NUM_GROUPS, M / M_TILE, 1);     // x = group (fast) -> H tile L2 reuse
    hlr_fused_kernel<<<grid, 256, 0, stream>>>(H, VT, U, Out);
}